// BipartiteGATv2Convolution_40200893891317
// MI455X (gfx1250) — compile-verified
//
#include <hip/hip_runtime.h>
#include <hip/hip_bf16.h>

typedef __attribute__((ext_vector_type(2))) float v2f;
typedef __attribute__((ext_vector_type(8))) float v8f;

// D = A(16x4) * B(4x16) + C, fp32 WMMA. A: lanes0-15 K={k,k+1}, lanes16-31 K={k+2,k+3};
// B mirrored; C/D: VGPR v holds M=v+8*(lane/16), N=lane%16.
#define WMMA_F32(a,b,c) __builtin_amdgcn_wmma_f32_16x16x4_f32(false,(a),false,(b),(short)0,(c),false,false)

// ---- CDNA5 async global->LDS copy (ASYNCcnt path, no VGPR round-trip) ----
// saddr form: global_load_async_to_lds_b128 vdst(LDS byte addr), vaddr(32b byte off), saddr(base)
static __device__ __forceinline__ void async_b128(unsigned lds_byte, unsigned gbl_byte,
                                                  const void* base) {
  asm volatile("global_load_async_to_lds_b128 %0, %1, %2"
               :: "v"(lds_byte), "v"(gbl_byte), "s"(base)
               : "memory");
}
static __device__ __forceinline__ void wait_async0() {
  asm volatile("s_wait_asynccnt 0x0" ::: "memory");
}
// generic (flat) address of a __shared__ object truncated to 32 bits == LDS byte offset
static __device__ __forceinline__ unsigned lds_addr32(const void* p) {
  return (unsigned)(unsigned long long)p;
}

// ---- order-preserving float <-> uint for atomic max on signed floats ----
static __device__ __forceinline__ unsigned fkey(float x) {
  unsigned u = __float_as_uint(x);
  return (u & 0x80000000u) ? ~u : (u | 0x80000000u);
}
static __device__ __forceinline__ float fdec(unsigned k) {
  return __uint_as_float((k & 0x80000000u) ? (k & 0x7FFFFFFFu) : ~k);
}

__global__ void zero_k(unsigned* __restrict__ p, size_t n) {
  size_t i = (size_t)blockIdx.x * blockDim.x + threadIdx.x;
  if (i < n) p[i] = 0u;
}

// ---------------- C[M,128] = A[M,K] @ B[K,128] + bias (optional relu) ----------------
// 256 threads = 8 waves; wave owns a 16-row stripe; B staged in LDS (async DMA) in
// 128-row chunks; A fragments live in registers across all 8 N-tiles (A read once).
template<int K, bool RELU>
__global__ __launch_bounds__(256) void gemm_k(const float* __restrict__ A,
                                              const float* __restrict__ B,
                                              const float* __restrict__ bias,
                                              float* __restrict__ C, int M) {
  __shared__ float Bs[128 * 128];  // 64 KB of the 320 KB WGP LDS
  const int wave = threadIdx.x >> 5;
  const int lane = threadIdx.x & 31;
  const int half = lane >> 4;
  const int ln16 = lane & 15;
  const int m0 = (blockIdx.x * 8 + wave) * 16;
  const int row = m0 + ln16;
  const bool rowok = row < M;
  const unsigned lds0 = lds_addr32(Bs);

  v8f acc[8];
#pragma unroll
  for (int nt = 0; nt < 8; ++nt) acc[nt] = (v8f){};

  for (int kc = 0; kc < K; kc += 128) {
    __syncthreads();
    {
      const void* bsrc = (const void*)(B + (size_t)kc * 128);
      // 64 KB = 4096 x 16B; 256 threads x 16 async b128 ops
      for (int i = threadIdx.x; i < 128 * 128 / 4; i += 256)
        async_b128(lds0 + (unsigned)i * 16u, (unsigned)i * 16u, bsrc);
      wait_async0();
    }
    __syncthreads();

    v2f a[32];
    if (rowok) {
      const float* ap = A + (size_t)row * K + kc;
#pragma unroll
      for (int ks = 0; ks < 32; ++ks) {
        int kb = ks * 4 + half * 2;
        a[ks].x = ap[kb];
        a[ks].y = ap[kb + 1];
      }
    } else {
#pragma unroll
      for (int ks = 0; ks < 32; ++ks) { a[ks].x = 0.f; a[ks].y = 0.f; }
    }
#pragma unroll
    for (int nt = 0; nt < 8; ++nt) {
      const float* bs = Bs + nt * 16 + ln16;
#pragma unroll
      for (int ks = 0; ks < 32; ++ks) {
        int kb = ks * 4 + half * 2;
        v2f b;
        b.x = bs[(size_t)kb * 128];
        b.y = bs[(size_t)(kb + 1) * 128];
        acc[nt] = WMMA_F32(a[ks], b, acc[nt]);
      }
    }
  }
#pragma unroll
  for (int nt = 0; nt < 8; ++nt) {
    int n = nt * 16 + ln16;
    float bv = bias[n];
#pragma unroll
    for (int v = 0; v < 8; ++v) {
      int r = m0 + v + 8 * half;
      if (r < M) {
        float x = acc[nt][v] + bv;
        if (RELU) x = fmaxf(x, 0.f);
        C[(size_t)r * 128 + n] = x;
      }
    }
  }
}

// ---------------- fused: e = EF@We (WMMA, never materialized) ; z = leakyrelu(e + xl[src] + xr[dst])
// logits[e,h] = z . att[h]; atomic segment-max keys per (dst,head) ----------------
__global__ __launch_bounds__(256) void edge_logits_k(
    const float* __restrict__ EF, const float* __restrict__ We,
    const int* __restrict__ srcI, const int* __restrict__ dstI,
    const float* __restrict__ xl, const float* __restrict__ xr,
    const float* __restrict__ att, float* __restrict__ logits,
    unsigned* __restrict__ mkey, int E) {
  __shared__ float Ws[128 * 128];  // W_e, 64 KB, async-DMA'd once per block
  {
    const unsigned lds0 = lds_addr32(Ws);
    for (int i = threadIdx.x; i < 128 * 128 / 4; i += 256)
      async_b128(lds0 + (unsigned)i * 16u, (unsigned)i * 16u, (const void*)We);
    wait_async0();
  }
  __syncthreads();

  const int wave = threadIdx.x >> 5;
  const int lane = threadIdx.x & 31;
  const int half = lane >> 4;
  const int ln16 = lane & 15;
  const long e0 = ((long)blockIdx.x * 8 + wave) * 16;  // 16 edges per wave
  long erow = e0 + ln16; if (erow > (long)E - 1) erow = (long)E - 1;

  // A fragments: edge_features rows, read exactly once, reused for all 8 N-tiles
  v2f a[32];
  {
    const float* ap = EF + (size_t)erow * 128;
#pragma unroll
    for (int ks = 0; ks < 32; ++ks) {
      int kb = ks * 4 + half * 2;
      a[ks].x = ap[kb];
      a[ks].y = ap[kb + 1];
    }
  }
  const int s_reg = srcI[erow];  // lane l and l+16 hold indices of edge e0+(l&15)
  const int d_reg = dstI[erow];

#pragma unroll
  for (int h = 0; h < 4; ++h) {      // head h covers channels 32h..32h+31 = 2 N-tiles
    float part[8];
#pragma unroll
    for (int v = 0; v < 8; ++v) part[v] = 0.f;
#pragma unroll
    for (int sub = 0; sub < 2; ++sub) {
      const int nt = h * 2 + sub;
      v8f acc = (v8f){};
      const float* bs = Ws + nt * 16 + ln16;
#pragma unroll
      for (int ks = 0; ks < 32; ++ks) {
        int kb = ks * 4 + half * 2;
        v2f b;
        b.x = bs[(size_t)kb * 128];
        b.y = bs[(size_t)(kb + 1) * 128];
        acc = WMMA_F32(a[ks], b, acc);
      }
      const int n = nt * 16 + ln16;     // output channel this lane holds
      const float av = att[n];          // att flat [H*O] matches channel index
#pragma unroll
      for (int v = 0; v < 8; ++v) {     // VGPR v -> edge m = v + 8*half
        int m = v + 8 * half;
        int si = __shfl(s_reg, m, 32);
        int di = __shfl(d_reg, m, 32);
        float z = acc[v] + xl[(size_t)si * 128 + n] + xr[(size_t)di * 128 + n];
        z = (z > 0.f) ? z : 0.2f * z;   // leaky_relu(0.2)
        part[v] += z * av;
      }
    }
    // sum across the 16 lanes of each half (channels of this head)
#pragma unroll
    for (int v = 0; v < 8; ++v) {
      float p = part[v];
      p += __shfl_xor(p, 1, 32);
      p += __shfl_xor(p, 2, 32);
      p += __shfl_xor(p, 4, 32);
      p += __shfl_xor(p, 8, 32);
      part[v] = p;
    }
    if (ln16 == 0) {  // lanes 0 and 16 write edges m=v and m=8+v
#pragma unroll
      for (int v = 0; v < 8; ++v) {
        long e = e0 + v + 8 * half;
        if (e < E) {
          float lg = part[v];
          logits[e * 4 + h] = lg;
          atomicMax(&mkey[(size_t)dstI[e] * 4 + h], fkey(lg));
        }
      }
    }
  }
}

// ---------------- ex = exp(logit - m[dst]); den[dst,h] += ex ----------------
__global__ void softmax_norm_k(const int* __restrict__ dstI,
                               float* __restrict__ logits,
                               const unsigned* __restrict__ mkey,
                               float* __restrict__ den, long E4) {
  long i = (long)blockIdx.x * blockDim.x + threadIdx.x;
  if (i >= E4) return;
  long e = i >> 2; int h = (int)(i & 3);
  int d = dstI[e];
  float m = fdec(mkey[(size_t)d * 4 + h]);
  float ex = expf(logits[i] - m);
  logits[i] = ex;  // reuse buffer for exp values
  atomicAdd(&den[(size_t)d * 4 + h], ex);
}

// ---------------- msg[dst] += alpha * xl[src] (wave per edge, lane = 4 channels) ----------------
__global__ __launch_bounds__(256) void scatter_msg_k(
    const int* __restrict__ srcI, const int* __restrict__ dstI,
    const float* __restrict__ ex, const float* __restrict__ den,
    const float* __restrict__ xl, float* __restrict__ msg, long E) {
  long e = (long)blockIdx.x * 8 + (threadIdx.x >> 5);
  if (e >= E) return;
  int lane = threadIdx.x & 31;
  int h = lane >> 3;  // channels 4*lane..4*lane+3 belong to head lane/8
  int s = srcI[e], d = dstI[e];
  float alpha = ex[e * 4 + h] / (den[(size_t)d * 4 + h] + 1e-16f);
  float4 xv = *(const float4*)(xl + (size_t)s * 128 + lane * 4);
  float* md = msg + (size_t)d * 128 + lane * 4;
  atomicAdd(md + 0, xv.x * alpha);
  atomicAdd(md + 1, xv.y * alpha);
  atomicAdd(md + 2, xv.z * alpha);
  atomicAdd(md + 3, xv.w * alpha);
}

// ---------------- layernorm(msg + b_conv) || right -> hcat[NR,256] ----------------
__global__ __launch_bounds__(256) void ln_concat_k(
    const float* __restrict__ msg, const float* __restrict__ bconv,
    const float* __restrict__ g, const float* __restrict__ b,
    const float* __restrict__ right, float* __restrict__ hcat, int NR) {
  int r = blockIdx.x * 8 + (threadIdx.x >> 5);
  if (r >= NR) return;
  int lane = threadIdx.x & 31;
  int c = lane * 4;
  float4 v = *(const float4*)(msg + (size_t)r * 128 + c);
  float4 bc = *(const float4*)(bconv + c);
  v.x += bc.x; v.y += bc.y; v.z += bc.z; v.w += bc.w;
  float s = v.x + v.y + v.z + v.w;
  s += __shfl_xor(s, 1, 32);  s += __shfl_xor(s, 2, 32);
  s += __shfl_xor(s, 4, 32);  s += __shfl_xor(s, 8, 32);  s += __shfl_xor(s, 16, 32);
  float mu = s * (1.f / 128.f);
  float dx = v.x - mu, dy = v.y - mu, dz = v.z - mu, dw = v.w - mu;
  float q = dx * dx + dy * dy + dz * dz + dw * dw;
  q += __shfl_xor(q, 1, 32);  q += __shfl_xor(q, 2, 32);
  q += __shfl_xor(q, 4, 32);  q += __shfl_xor(q, 8, 32);  q += __shfl_xor(q, 16, 32);
  float rs = rsqrtf(q * (1.f / 128.f) + 1e-5f);
  float4 gg = *(const float4*)(g + c);
  float4 bb = *(const float4*)(b + c);
  float4 o;
  o.x = dx * rs * gg.x + bb.x;  o.y = dy * rs * gg.y + bb.y;
  o.z = dz * rs * gg.z + bb.z;  o.w = dw * rs * gg.w + bb.w;
  *(float4*)(hcat + (size_t)r * 256 + c) = o;
  *(float4*)(hcat + (size_t)r * 256 + 128 + c) =
      *(const float4*)(right + (size_t)r * 128 + c);
}

extern "C" void kernel_launch(void* const* d_in, const int* in_sizes, int n_in,
                              void* d_out, int out_size, void* d_ws, size_t ws_size,
                              hipStream_t stream) {
  (void)n_in; (void)out_size; (void)ws_size;
  const float* left   = (const float*)d_in[0];
  const int*   eidx   = (const int*)d_in[1];
  const float* ef     = (const float*)d_in[2];
  const float* right  = (const float*)d_in[3];
  const float* W_l    = (const float*)d_in[4];
  const float* b_l    = (const float*)d_in[5];
  const float* W_r    = (const float*)d_in[6];
  const float* b_r    = (const float*)d_in[7];
  const float* W_e    = (const float*)d_in[8];
  const float* att    = (const float*)d_in[9];
  const float* b_conv = (const float*)d_in[10];
  const float* ln_g   = (const float*)d_in[11];
  const float* ln_b   = (const float*)d_in[12];
  const float* W1     = (const float*)d_in[13];
  const float* b1     = (const float*)d_in[14];
  const float* W2     = (const float*)d_in[15];
  const float* b2     = (const float*)d_in[16];

  const int NL = in_sizes[0] / 128;
  const int E  = in_sizes[1] / 2;
  const int NR = in_sizes[3] / 128;
  const int* srcI = eidx;
  const int* dstI = eidx + E;

  // workspace carve-out (256B aligned)
  char* ws = (char*)d_ws;
  auto carve = [&](size_t bytes) {
    char* p = ws; ws += (bytes + 255) & ~(size_t)255; return p;
  };
  float*    xl     = (float*)carve((size_t)NL * 128 * 4);
  float*    xr     = (float*)carve((size_t)NR * 128 * 4);
  float*    logits = (float*)carve((size_t)E * 4 * 4);
  unsigned* mkey   = (unsigned*)carve((size_t)NR * 4 * 4);
  float*    den    = (float*)carve((size_t)NR * 4 * 4);
  float*    msg    = (float*)carve((size_t)NR * 128 * 4);
  float*    hcat   = (float*)carve((size_t)NR * 256 * 4);
  float*    h1     = (float*)carve((size_t)NR * 128 * 4);

  // init accumulators (key 0 == below any real float's order key)
  size_t nz = (size_t)NR * 4;
  zero_k<<<(unsigned)((nz + 255) / 256), 256, 0, stream>>>(mkey, nz);
  zero_k<<<(unsigned)((nz + 255) / 256), 256, 0, stream>>>((unsigned*)den, nz);
  size_t nzm = (size_t)NR * 128;
  zero_k<<<(unsigned)((nzm + 255) / 256), 256, 0, stream>>>((unsigned*)msg, nzm);

  // node projections
  gemm_k<128, false><<<(NL + 127) / 128, 256, 0, stream>>>(left,  W_l, b_l, xl, NL);
  gemm_k<128, false><<<(NR + 127) / 128, 256, 0, stream>>>(right, W_r, b_r, xr, NR);

  // fused edge GEMM + attention logits + segment max
  edge_logits_k<<<(E + 127) / 128, 256, 0, stream>>>(ef, W_e, srcI, dstI, xl, xr,
                                                     att, logits, mkey, E);
  // softmax
  long E4 = (long)E * 4;
  softmax_norm_k<<<(unsigned)((E4 + 255) / 256), 256, 0, stream>>>(dstI, logits, mkey, den, E4);
  // weighted message scatter
  scatter_msg_k<<<(unsigned)(((long)E + 7) / 8), 256, 0, stream>>>(srcI, dstI, logits, den,
                                                                   xl, msg, (long)E);
  // layernorm + concat
  ln_concat_k<<<(NR + 7) / 8, 256, 0, stream>>>(msg, b_conv, ln_g, ln_b, right, hcat, NR);
  // output MLP
  gemm_k<256, true ><<<(NR + 127) / 128, 256, 0, stream>>>(hcat, W1, b1, h1, NR);
  gemm_k<128, false><<<(NR + 127) / 128, 256, 0, stream>>>(h1,   W2, b2, (float*)d_out, NR);
}